// SmallestAdditionTransformer_37752762531957
// MI455X (gfx1250) — compile-verified
//
#include <hip/hip_runtime.h>
#include <math.h>

// Tiny transformer, fully fused: one wave32 per sequence.
// WMMA f32 16x16x4 for attention scores, attn@V, and the vocab head;
// VALU only for the d=6 layernorms / tiny projections / GELU;
// branchless vectorized softmax (b128 LDS traffic, register-masked causal).

#define T      34
#define D      6
#define TOKD   3
#define POSD   3
#define HD     3
#define NH     2
#define FFND   6
#define VOC    14
#define EPSV   1e-5f

#define WAVES  4    // waves per block (1 sequence per wave)
#define TQ     48   // q/k rows padded to 3 WMMA tiles
#define TV     40   // v K-rows padded (36 used)
#define SROW   40   // score-buffer row stride (cols 0..35 used)

typedef __attribute__((ext_vector_type(2))) float v2f;
typedef __attribute__((ext_vector_type(8))) float v8f;

__device__ __forceinline__ void layernorm6(const float* x, const float* g,
                                           const float* b, float* o) {
  float mu = 0.f;
  for (int i = 0; i < 6; ++i) mu += x[i];
  mu *= (1.f / 6.f);
  float var = 0.f;
  for (int i = 0; i < 6; ++i) { float d = x[i] - mu; var += d * d; }
  var *= (1.f / 6.f);
  float r = rsqrtf(var + EPSV);
  for (int i = 0; i < 6; ++i) o[i] = (x[i] - mu) * r * g[i] + b[i];
}

__global__ __launch_bounds__(32 * WAVES)
void tinytx_kernel(const int* __restrict__ idx,
                   const float* __restrict__ tok_emb,
                   const float* __restrict__ pos_enc,
                   const float* __restrict__ wq,
                   const float* __restrict__ wk,
                   const float* __restrict__ wv,
                   const float* __restrict__ wo,
                   const float* __restrict__ ln1_g, const float* __restrict__ ln1_b,
                   const float* __restrict__ ln2_g, const float* __restrict__ ln2_b,
                   const float* __restrict__ w1,   const float* __restrict__ b1,
                   const float* __restrict__ w2,   const float* __restrict__ b2,
                   const float* __restrict__ lnf_g, const float* __restrict__ lnf_b,
                   const float* __restrict__ w_head,
                   float* __restrict__ out) {
  __shared__ float xbuf[WAVES][T][8];          // residual stream
  __shared__ float obuf[WAVES][T][8];          // attention head outputs
  __shared__ float qbuf[WAVES][NH][TQ][4];     // q (later reused as LNf output)
  __shared__ float kbuf[WAVES][NH][TQ][4];     // k
  __shared__ float vbuf[WAVES][NH][TV][4];     // v (col 3 kept zero)
  __shared__ float sbuf[WAVES][TQ + 1][SROW];  // scores/probs; row 48 = dump slot
  __shared__ float invbuf[WAVES][TQ];          // 1/softmax_sum per row

  const int wave = threadIdx.x >> 5;
  const int lane = threadIdx.x & 31;
  const int hl   = lane >> 4;   // 16-lane half
  const int ln16 = lane & 15;
  const long seq = (long)blockIdx.x * WAVES + wave;

  float (*xb)[8]     = xbuf[wave];
  float (*ob)[8]     = obuf[wave];
  float (*qb)[TQ][4] = qbuf[wave];
  float (*kb)[TQ][4] = kbuf[wave];
  float (*vb)[TV][4] = vbuf[wave];
  float (*sb)[SROW]  = sbuf[wave];
  float* invb        = invbuf[wave];
  float* dump        = &sbuf[wave][TQ][0];     // 40-float scratch row (16B aligned)

  // ---- zero-init padded regions (pad lanes contribute exact zeros) ----
  for (int r = lane; r < TQ; r += 32) {
    for (int h = 0; h < NH; ++h)
      for (int c = 0; c < 4; ++c) { qb[h][r][c] = 0.f; kb[h][r][c] = 0.f; }
    if (r < TV)
      for (int h = 0; h < NH; ++h)
        for (int c = 0; c < 4; ++c) vb[h][r][c] = 0.f;
    if (r >= T)
      for (int c = 0; c < SROW; ++c) sb[r][c] = 0.f;   // pad rows read by av A-tiles
  }

  // ---- embed: x = concat(tok_emb[idx], pos_enc) ----
  const int* myidx = idx + seq * T;
  for (int t = lane; t < T; t += 32) {
    int id = myidx[t];
    for (int c = 0; c < TOKD; ++c) xb[t][c]        = tok_emb[id * TOKD + c];
    for (int c = 0; c < POSD; ++c) xb[t][TOKD + c] = pos_enc[t * POSD + c];
  }

  // ---- LN1 + QKV projections (lane per token) ----
  for (int t = lane; t < T; t += 32) {
    float xl[6], h[6];
    for (int i = 0; i < 6; ++i) xl[i] = xb[t][i];
    layernorm6(xl, ln1_g, ln1_b, h);
    for (int hh = 0; hh < NH; ++hh)
      for (int c = 0; c < HD; ++c) {
        float q = 0.f, k = 0.f, v = 0.f;
        for (int p = 0; p < POSD; ++p) {
          q += h[3 + p] * wq[p * (NH * HD) + hh * HD + c];
          k += h[3 + p] * wk[p * (NH * HD) + hh * HD + c];
        }
        for (int p = 0; p < TOKD; ++p)
          v += h[p] * wv[p * (NH * HD) + hh * HD + c];
        qb[hh][t][c] = q;
        kb[hh][t][c] = k;
        vb[hh][t][c] = v;
      }
  }

  const float scale = 0.57735026919f;  // 1/sqrt(HEAD_DIM)
  const float NEGBIG = -3.0e38f;

  // ---- attention, per head ----
  for (int hh = 0; hh < NH; ++hh) {
    // scores: S = (q @ k^T) * scale, 16x16 tiles, causal -> keep nt <= mt only
    for (int mt = 0; mt < 3; ++mt)
      for (int nt = 0; nt <= mt; ++nt) {
        v2f a, b;
        int M = mt * 16 + ln16;               // q row (same for both halves)
        a.x = qb[hh][M][2 * hl + 0];          // A: lanes0-15 K={0,1}, 16-31 K={2,3}
        a.y = qb[hh][M][2 * hl + 1];
        int N = nt * 16 + ln16;               // k row (score column)
        b.x = kb[hh][N][2 * hl + 0];
        b.y = kb[hh][N][2 * hl + 1];
        v8f c = {};
        c = __builtin_amdgcn_wmma_f32_16x16x4_f32(false, a, false, b,
                                                  (short)0, c, false, false);
        int col = nt * 16 + ln16;
        for (int r = 0; r < 8; ++r) {
          int row = mt * 16 + r + 8 * hl;
          float* p = (row < T && col < 36) ? &sb[row][col] : &dump[lane];
          *p = c[r] * scale;                  // pre-scale; dump slot if padded
        }
      }

    // softmax: lane per row, fully vectorized (9x b128 in, 9x b128 out),
    // causal mask in registers; exp(-huge)==0 doubles as the zero-fill.
    // 1/sum is applied later at the av epilogue.
    for (int q0 = 0; q0 < T; q0 += 32) {
      int q = q0 + lane;
      bool act = q < T;
      float* rp = act ? &sb[q][0] : dump;     // inactive lanes shadow-work on dump
      float rv[36];
      for (int c = 0; c < 36; c += 4) {
        float4 t4 = *(const float4*)(rp + c);
        rv[c + 0] = t4.x; rv[c + 1] = t4.y; rv[c + 2] = t4.z; rv[c + 3] = t4.w;
      }
      float mx = NEGBIG;
      for (int k = 0; k < 36; ++k) {
        rv[k] = (k <= q) ? rv[k] : NEGBIG;    // k=0 always valid => mx finite
        mx = fmaxf(mx, rv[k]);
      }
      float sum = 0.f;
      for (int k = 0; k < 36; ++k) {
        float e = __expf(rv[k] - mx);         // masked slots -> exactly 0
        rv[k] = e;
        sum += e;
      }
      invb[act ? q : TQ - 1] = 1.f / sum;
      for (int c = 0; c < 36; c += 4) {
        float4 t4;
        t4.x = rv[c + 0]; t4.y = rv[c + 1]; t4.z = rv[c + 2]; t4.w = rv[c + 3];
        *(float4*)(rp + c) = t4;
      }
    }

    // attn @ v : 3 M-tiles, chained accumulation over 9 K-steps
    int cc = ln16 < 4 ? ln16 : 3;             // col 3 of vbuf is zero => pad lanes get 0
    for (int mt = 0; mt < 3; ++mt) {
      v8f c = {};
      for (int kk = 0; kk < 36; kk += 4) {
        v2f a, b;
        int row = mt * 16 + ln16;
        a.x = sb[row][kk + 2 * hl + 0];
        a.y = sb[row][kk + 2 * hl + 1];
        b.x = vb[hh][kk + 2 * hl + 0][cc];
        b.y = vb[hh][kk + 2 * hl + 1][cc];
        c = __builtin_amdgcn_wmma_f32_16x16x4_f32(false, a, false, b,
                                                  (short)0, c, false, false);
      }
      for (int r = 0; r < 8; ++r) {
        int row = mt * 16 + r + 8 * hl;       // row <= 47 < TQ
        float* p = (row < T && ln16 < HD) ? &ob[row][hh * HD + ln16] : &dump[lane];
        *p = c[r] * invb[row];
      }
    }
  }

  // ---- tail: WO + residual, LN2, FFN (exact-erf GELU), LNf -> stage to LDS ----
  float (*gb)[8] = (float(*)[8]) & qbuf[wave][0][0][0];  // reuse q staging: 48x8 floats
  for (int t = lane; t < T; t += 32) {
    float xv[6], att[6];
    for (int i = 0; i < 6; ++i) { xv[i] = xb[t][i]; att[i] = ob[t][i]; }
    for (int j = 0; j < 6; ++j) {
      float s = 0.f;
      for (int i = 0; i < 6; ++i) s += att[i] * wo[i * 6 + j];
      xv[j] += s;
    }
    float h[6];
    layernorm6(xv, ln2_g, ln2_b, h);
    float f[6];
    for (int j = 0; j < FFND; ++j) {
      float s = b1[j];
      for (int i = 0; i < 6; ++i) s += h[i] * w1[i * FFND + j];
      f[j] = 0.5f * s * (1.0f + erff(s * 0.70710678118f));
    }
    for (int j = 0; j < 6; ++j) {
      float s = b2[j];
      for (int i = 0; i < FFND; ++i) s += f[i] * w2[i * 6 + j];
      xv[j] += s;
    }
    float g[6];
    layernorm6(xv, lnf_g, lnf_b, g);
    for (int i = 0; i < 6; ++i) gb[t][i] = g[i];
    gb[t][6] = 0.f;
    gb[t][7] = 0.f;
  }
  for (int r = T + lane; r < TQ; r += 32)      // zero pad rows of LNf staging
    for (int c = 0; c < 8; ++c) gb[r][c] = 0.f;

  // ---- vocab head via WMMA: logits = g(48x8) @ w_head(8x16), direct to HBM ----
  v2f hbt[2];
  for (int kc = 0; kc < 2; ++kc) {
    int kr0 = kc * 4 + 2 * hl, kr1 = kr0 + 1;
    int nn = ln16 < VOC ? ln16 : VOC - 1;
    float bx = w_head[(kr0 < D ? kr0 : D - 1) * VOC + nn];
    float by = w_head[(kr1 < D ? kr1 : D - 1) * VOC + nn];
    hbt[kc].x = (kr0 < D && ln16 < VOC) ? bx : 0.f;
    hbt[kc].y = (kr1 < D && ln16 < VOC) ? by : 0.f;
  }
  float* obase = out + (long)seq * T * VOC;
  for (int mt = 0; mt < 3; ++mt) {
    v8f c = {};
    for (int kc = 0; kc < 2; ++kc) {
      v2f a;
      int M = mt * 16 + ln16;
      a.x = gb[M][kc * 4 + 2 * hl + 0];
      a.y = gb[M][kc * 4 + 2 * hl + 1];
      c = __builtin_amdgcn_wmma_f32_16x16x4_f32(false, a, false, hbt[kc],
                                                (short)0, c, false, false);
    }
    if (ln16 < VOC) {
      for (int r = 0; r < 8; ++r) {
        int row = mt * 16 + r + 8 * hl;
        if (row < T) obase[row * VOC + ln16] = c[r];
      }
    }
  }
}

extern "C" void kernel_launch(void* const* d_in, const int* in_sizes, int n_in,
                              void* d_out, int out_size, void* d_ws, size_t ws_size,
                              hipStream_t stream) {
  const int*   idx     = (const int*)d_in[0];
  const float* tok_emb = (const float*)d_in[1];
  const float* pos_enc = (const float*)d_in[2];
  const float* wq      = (const float*)d_in[3];
  const float* wk      = (const float*)d_in[4];
  const float* wv      = (const float*)d_in[5];
  const float* wo      = (const float*)d_in[6];
  const float* ln1_g   = (const float*)d_in[7];
  const float* ln1_b   = (const float*)d_in[8];
  const float* ln2_g   = (const float*)d_in[9];
  const float* ln2_b   = (const float*)d_in[10];
  const float* w1      = (const float*)d_in[11];
  const float* b1      = (const float*)d_in[12];
  const float* w2      = (const float*)d_in[13];
  const float* b2      = (const float*)d_in[14];
  const float* lnf_g   = (const float*)d_in[15];
  const float* lnf_b   = (const float*)d_in[16];
  const float* w_head  = (const float*)d_in[17];

  int Bn = in_sizes[0] / T;                 // 65536 sequences
  int blocks = (Bn + WAVES - 1) / WAVES;    // 1 sequence per wave
  tinytx_kernel<<<blocks, 32 * WAVES, 0, stream>>>(
      idx, tok_emb, pos_enc, wq, wk, wv, wo, ln1_g, ln1_b, ln2_g, ln2_b,
      w1, b1, w2, b2, lnf_g, lnf_b, w_head, (float*)d_out);
}